// GCNnet_40407052321143
// MI455X (gfx1250) — compile-verified
//
#include <hip/hip_runtime.h>
#include <hip/hip_bf16.h>

typedef __attribute__((ext_vector_type(2))) float v2f;
typedef __attribute__((ext_vector_type(8))) float v8f;

#define F_IN 256
#define H1   32
#define C2   16

// Guaranteed-native fp32 scatter add: no-return global_atomic_add_f32 at device
// scope (L2 atomic units). Bypasses clang's CAS-loop legalization entirely.
__device__ __forceinline__ void atomic_add_f32(float* p, float v) {
#if defined(__gfx1250__)
    asm volatile("global_atomic_add_f32 %0, %1, off scope:SCOPE_DEV"
                 :: "v"(p), "v"(v)
                 : "memory");
#else
    atomicAdd(p, v);
#endif
}

// ---- init: zero scatter targets, zero integer degree counters ----
__global__ void k_init(unsigned* __restrict__ degc, float* __restrict__ agg1,
                       float* __restrict__ agg2, int N) {
    long long t = (long long)blockIdx.x * blockDim.x + threadIdx.x;
    if (t < (long long)N * H1) agg1[t] = 0.0f;
    if (t < (long long)N * C2) agg2[t] = 0.0f;
    if (t < N)                 degc[t] = 0u;
}

// ---- degree count over edge destinations (native u32 atomics) ----
__global__ void k_deg(const int* __restrict__ dst, unsigned* __restrict__ degc, int E) {
    int t = blockIdx.x * blockDim.x + threadIdx.x;
    if (t < E) atomicAdd(&degc[dst[t]], 1u);
}

// ---- dinv = rsqrt(deg_count + 1 self loop), in place (int -> float) ----
__global__ void k_dinv(float* __restrict__ buf, int N) {
    int t = blockIdx.x * blockDim.x + threadIdx.x;
    if (t < N) {
        unsigned c = ((const unsigned*)buf)[t];
        buf[t] = rsqrtf((float)(c + 1u));
    }
}

// ---- GEMM1: hs1[i,:] = dinv[i] * (x[i,:] @ W1)   [N,256]x[256,32] ----
// One wave -> 16x16 output tile via V_WMMA_F32_16X16X4_F32, K-loop of 64.
__global__ void __launch_bounds__(128) k_gemm1(const float* __restrict__ x,
                                               const float* __restrict__ W1,
                                               const float* __restrict__ dinv,
                                               float* __restrict__ hs1, int N) {
    __shared__ float lw[F_IN * H1];           // 32 KB: whole W1
    const int tid = threadIdx.x;
    {   // cooperative 128-bit LDS fill
        const float4* s4 = (const float4*)W1;
        float4* d4 = (float4*)lw;
        for (int i = tid; i < (F_IN * H1) / 4; i += 128) d4[i] = s4[i];
    }
    __syncthreads();

    const int lane = tid & 31;
    const int waveGlobal = blockIdx.x * 4 + (tid >> 5);
    const int rowTile = (waveGlobal >> 1) * 16;
    const int colTile = (waveGlobal & 1) * 16;
    if (rowTile + 16 > N) return;             // wave-uniform guard, EXEC stays full

    const int l15  = lane & 15;
    const int koff = (lane >> 4) * 2;         // lanes 0-15 -> K0/K1, 16-31 -> K2/K3
    const float* xrow = x + (size_t)(rowTile + l15) * F_IN + koff;

    v8f acc = {};
    #pragma unroll 8
    for (int k = 0; k < F_IN; k += 4) {
        v2f a = *(const v2f*)(xrow + k);      // A: 16x4 fragment
        v2f b;                                // B: 4x16 fragment from LDS
        int bidx = (k + koff) * H1 + colTile + l15;
        b.x = lw[bidx];
        b.y = lw[bidx + H1];
        acc = __builtin_amdgcn_wmma_f32_16x16x4_f32(false, a, false, b,
                                                    (short)0, acc, false, false);
    }

    const int mBase = rowTile + (lane >> 4) * 8;
    const int col   = colTile + l15;
    #pragma unroll
    for (int v = 0; v < 8; v++) {
        int row = mBase + v;
        hs1[(size_t)row * H1 + col] = acc[v] * dinv[row];
    }
}

// ---- GEMM2: hs2[i,:] = dinv[i] * (h[i,:] @ W2)   [N,32]x[32,16] ----
__global__ void __launch_bounds__(128) k_gemm2(const float* __restrict__ h,
                                               const float* __restrict__ W2,
                                               const float* __restrict__ dinv,
                                               float* __restrict__ hs2, int N) {
    __shared__ float lw[H1 * C2];             // 2 KB: whole W2
    const int tid = threadIdx.x;
    for (int i = tid; i < H1 * C2; i += 128) lw[i] = W2[i];
    __syncthreads();

    const int lane = tid & 31;
    const int waveGlobal = blockIdx.x * 4 + (tid >> 5);
    const int rowTile = waveGlobal * 16;
    if (rowTile + 16 > N) return;

    const int l15  = lane & 15;
    const int koff = (lane >> 4) * 2;
    const float* hrow = h + (size_t)(rowTile + l15) * H1 + koff;

    v8f acc = {};
    #pragma unroll
    for (int k = 0; k < H1; k += 4) {
        v2f a = *(const v2f*)(hrow + k);
        v2f b;
        int bidx = (k + koff) * C2 + l15;
        b.x = lw[bidx];
        b.y = lw[bidx + C2];
        acc = __builtin_amdgcn_wmma_f32_16x16x4_f32(false, a, false, b,
                                                    (short)0, acc, false, false);
    }

    const int mBase = rowTile + (lane >> 4) * 8;
    #pragma unroll
    for (int v = 0; v < 8; v++) {
        int row = mBase + v;
        hs2[(size_t)row * C2 + l15] = acc[v] * dinv[row];
    }
}

// ---- scatter layer 1: one wave per edge, 32 lanes = 32 contiguous features ----
__global__ void k_scatter32(const int* __restrict__ src, const int* __restrict__ dst,
                            const float* __restrict__ hs, float* __restrict__ agg,
                            int E) {
    long long t = (long long)blockIdx.x * blockDim.x + threadIdx.x;
    long long e = t >> 5;
    if (e >= E) return;
    int l = (int)(t & 31);
    int s = src[e], d = dst[e];
    atomic_add_f32(&agg[(size_t)d * H1 + l], hs[(size_t)s * H1 + l]);
}

// ---- scatter layer 2: 16 lanes per edge (2 edges per wave) ----
__global__ void k_scatter16(const int* __restrict__ src, const int* __restrict__ dst,
                            const float* __restrict__ hs, float* __restrict__ agg,
                            int E) {
    long long t = (long long)blockIdx.x * blockDim.x + threadIdx.x;
    long long e = t >> 4;
    if (e >= E) return;
    int l = (int)(t & 15);
    int s = src[e], d = dst[e];
    atomic_add_f32(&agg[(size_t)d * C2 + l], hs[(size_t)s * C2 + l]);
}

// ---- epilogue 1: out1 = relu(dinv*(agg1 + self hs1) + b1), in place on agg1 ----
__global__ void k_post1(float* __restrict__ agg1, const float* __restrict__ hs1,
                        const float* __restrict__ dinv, const float* __restrict__ b1,
                        long long n32) {
    long long t = (long long)blockIdx.x * blockDim.x + threadIdx.x;
    if (t >= n32) return;
    int row = (int)(t >> 5);
    int c   = (int)(t & 31);
    float v = dinv[row] * (agg1[t] + hs1[t]) + b1[c];
    agg1[t] = v > 0.0f ? v : 0.0f;
}

// ---- epilogue 2 + softmax over 16 classes, one thread per node ----
__global__ void k_post2_softmax(const float* __restrict__ agg2,
                                const float* __restrict__ hs2,
                                const float* __restrict__ dinv,
                                const float* __restrict__ b2,
                                float* __restrict__ out, int N) {
    int i = blockIdx.x * blockDim.x + threadIdx.x;
    if (i >= N) return;
    float di = dinv[i];
    float vals[C2];
    float m = -3.4e38f;
    #pragma unroll
    for (int c = 0; c < C2; c++) {
        float v = di * (agg2[(size_t)i * C2 + c] + hs2[(size_t)i * C2 + c]) + b2[c];
        vals[c] = v;
        m = fmaxf(m, v);
    }
    float s = 0.0f;
    #pragma unroll
    for (int c = 0; c < C2; c++) {
        float e = __expf(vals[c] - m);
        vals[c] = e;
        s += e;
    }
    float inv = 1.0f / s;
    #pragma unroll
    for (int c = 0; c < C2; c++) out[(size_t)i * C2 + c] = vals[c] * inv;
}

extern "C" void kernel_launch(void* const* d_in, const int* in_sizes, int n_in,
                              void* d_out, int out_size, void* d_ws, size_t ws_size,
                              hipStream_t stream) {
    const float* x  = (const float*)d_in[0];
    const int*   ei = (const int*)  d_in[1];
    const float* W1 = (const float*)d_in[2];
    const float* b1 = (const float*)d_in[3];
    const float* W2 = (const float*)d_in[4];
    const float* b2 = (const float*)d_in[5];
    float* out = (float*)d_out;

    const int N = in_sizes[0] / F_IN;
    const int E = in_sizes[1] / 2;
    const int* src = ei;        // edge_index[0]
    const int* dst = ei + E;    // edge_index[1]

    float* ws   = (float*)d_ws;
    float* deg  = ws;                         // N      (u32 counts -> dinv in place)
    float* hs1  = deg  + N;                   // N*32
    float* agg1 = hs1  + (size_t)N * H1;      // N*32   (becomes relu'd h1 in place)
    float* hs2  = agg1 + (size_t)N * H1;      // N*16
    float* agg2 = hs2  + (size_t)N * C2;      // N*16

    const int blk = 256;
    const long long n32 = (long long)N * H1;

    k_init<<<(unsigned)((n32 + blk - 1) / blk), blk, 0, stream>>>((unsigned*)deg, agg1, agg2, N);
    k_deg <<<(E + blk - 1) / blk, blk, 0, stream>>>(dst, (unsigned*)deg, E);
    k_dinv<<<(N + blk - 1) / blk, blk, 0, stream>>>(deg, N);

    int waves1 = (N / 16) * 2;                // 16x16 tiles over [N,32]
    k_gemm1<<<(waves1 + 3) / 4, 128, 0, stream>>>(x, W1, deg, hs1, N);

    long long t1 = (long long)E * 32;
    k_scatter32<<<(unsigned)((t1 + blk - 1) / blk), blk, 0, stream>>>(src, dst, hs1, agg1, E);
    k_post1<<<(unsigned)((n32 + blk - 1) / blk), blk, 0, stream>>>(agg1, hs1, deg, b1, n32);

    int waves2 = N / 16;                      // 16x16 tiles over [N,16]
    k_gemm2<<<(waves2 + 3) / 4, 128, 0, stream>>>(agg1, W2, deg, hs2, N);

    long long t2 = (long long)E * 16;
    k_scatter16<<<(unsigned)((t2 + blk - 1) / blk), blk, 0, stream>>>(src, dst, hs2, agg2, E);
    k_post2_softmax<<<(N + blk - 1) / blk, blk, 0, stream>>>(agg2, hs2, deg, b2, out, N);
}